// GCN2_12541304504853
// MI455X (gfx1250) — compile-verified
//
#include <hip/hip_runtime.h>
#include <math.h>

typedef __attribute__((ext_vector_type(2))) float v2f;
typedef __attribute__((ext_vector_type(8))) float v8f;

#define ALPHA_C   0.1f
#define IND       256
#define HD        96
#define OUTD      40
#define BN_EPS    1e-5f

// ---------------- degree / normalization ----------------
__global__ void k_deg_init(float* deg, int N) {
    int i = blockIdx.x * blockDim.x + threadIdx.x;
    if (i < N) deg[i] = 1.0f;                       // self-loop contribution
}
__global__ void k_deg_count(const int* __restrict__ col, float* deg, int E) {
    int i = blockIdx.x * blockDim.x + threadIdx.x;
    if (i < E) atomicAdd(&deg[col[i]], 1.0f);
}
__global__ void k_deg_rsqrt(float* deg, int N) {
    int i = blockIdx.x * blockDim.x + threadIdx.x;
    if (i < N) deg[i] = rsqrtf(deg[i]);             // deg >= 1 always (self-loops)
}

// ---------------- fc1: relu(X[N,256] @ W[256,96] + b) -> Y[N,96] ----------------
// Weights staged in LDS as K-pair-interleaved float2 so each lane's B operand
// (K = {2g, 2g+1}, same n) is ONE aligned ds_load_b64, no repacking movs.
__global__ __launch_bounds__(256)
void k_fc1(const float* __restrict__ X, const float* __restrict__ W,
           const float* __restrict__ bias, float* __restrict__ Y, int N) {
    __shared__ v2f sW2[64 * HD];                    // 48 KB: 128 K-rows per chunk
    const int tid  = threadIdx.x;
    const int wave = tid >> 5;
    const int lane = tid & 31;
    const int m    = lane & 15;
    const int g    = lane >> 4;                     // 0: K pair {0,1}, 1: K pair {2,3}

    int mtiles = (N + 15) >> 4;
    int mt = blockIdx.x * 8 + wave;
    if (mt >= mtiles) mt = mtiles - 1;              // clamp: keep EXEC all-1 at WMMA
    const int m0 = mt << 4;
    const int ar = min(m0 + m, N - 1);
    const float* Arow = X + (size_t)ar * IND;

    v8f acc[6];
    #pragma unroll
    for (int t = 0; t < 6; ++t) acc[t] = (v8f){0,0,0,0,0,0,0,0};

    for (int kc = 0; kc < IND; kc += 128) {
        __syncthreads();
        for (int i = tid; i < 64 * HD; i += 256) {
            int p = i / HD, n = i % HD;             // pair p covers K rows {2p, 2p+1}
            v2f w;
            w.x = W[(kc + 2 * p)     * HD + n];
            w.y = W[(kc + 2 * p + 1) * HD + n];
            sW2[i] = w;
        }
        __syncthreads();
        for (int k = 0; k < 128; k += 4) {
            const int kp = (k >> 1) + g;            // this lane's K pair index
            v2f a;
            a.x = Arow[kc + k + 2 * g];
            a.y = Arow[kc + k + 2 * g + 1];
            #pragma unroll
            for (int t = 0; t < 6; ++t) {
                v2f b = sW2[kp * HD + t * 16 + m];  // single ds_load_b64
                acc[t] = __builtin_amdgcn_wmma_f32_16x16x4_f32(
                    false, a, false, b, (short)0, acc[t], false, false);
            }
        }
    }
    #pragma unroll
    for (int t = 0; t < 6; ++t) {
        const int n  = t * 16 + m;
        const float bv = bias[n];
        #pragma unroll
        for (int r = 0; r < 8; ++r) {
            const int row = m0 + r + 8 * g;
            if (row < N) {
                float v = acc[t][r] + bv;
                Y[(size_t)row * HD + n] = v > 0.f ? v : 0.f;
            }
        }
    }
}

// ---------------- aggregation: h = (1-a)*D^-1/2 (A+I) D^-1/2 x + a*x0 ----------------
__global__ void k_agg_init(const float* __restrict__ x, const float* __restrict__ x0,
                           const float* __restrict__ dinv, float* __restrict__ h, int N) {
    long long idx = (long long)blockIdx.x * blockDim.x + threadIdx.x;
    if (idx < (long long)N * HD) {
        int n = (int)(idx / HD);
        float d = dinv[n];
        h[idx] = (1.0f - ALPHA_C) * d * d * x[idx] + ALPHA_C * x0[idx];
    }
}
__global__ void k_agg_edges(const int* __restrict__ ei, const float* __restrict__ dinv,
                            const float* __restrict__ x, float* __restrict__ h, int E) {
    int e    = (blockIdx.x * blockDim.x + threadIdx.x) >> 5;   // one wave per edge
    int lane = threadIdx.x & 31;
    if (e >= E) return;
    int r = ei[e], c = ei[E + e];
    float w = (1.0f - ALPHA_C) * dinv[r] * dinv[c];
    const float* xs = x + (size_t)r * HD;
    float* hd = h + (size_t)c * HD;
    __builtin_prefetch(xs + lane, 0, 0);            // global_prefetch_b8
    #pragma unroll
    for (int f = lane; f < HD; f += 32) atomicAdd(&hd[f], w * xs[f]);
}

// ---------------- conv: Y = (1-beta)*H + beta*(H[N,96] @ W[96,96]) ----------------
__global__ __launch_bounds__(256)
void k_conv(const float* __restrict__ H, const float* __restrict__ W,
            float* __restrict__ Y, int N, float beta) {
    __shared__ v2f sW2[48 * HD];                    // 36 KB, K-pair interleaved
    const int tid  = threadIdx.x;
    const int lane = tid & 31;
    const int m    = lane & 15;
    const int g    = lane >> 4;
    for (int i = tid; i < 48 * HD; i += 256) {
        int p = i / HD, n = i % HD;
        v2f w;
        w.x = W[(2 * p)     * HD + n];
        w.y = W[(2 * p + 1) * HD + n];
        sW2[i] = w;
    }
    __syncthreads();

    int mtiles = (N + 15) >> 4;
    int mt = blockIdx.x * 8 + (tid >> 5);
    if (mt >= mtiles) mt = mtiles - 1;
    const int m0 = mt << 4;
    const int ar = min(m0 + m, N - 1);
    const float* Arow = H + (size_t)ar * HD;

    v8f acc[6];
    #pragma unroll
    for (int t = 0; t < 6; ++t) acc[t] = (v8f){0,0,0,0,0,0,0,0};

    for (int k = 0; k < HD; k += 4) {
        const int kp = (k >> 1) + g;
        v2f a;
        a.x = Arow[k + 2 * g];
        a.y = Arow[k + 2 * g + 1];
        #pragma unroll
        for (int t = 0; t < 6; ++t) {
            v2f b = sW2[kp * HD + t * 16 + m];
            acc[t] = __builtin_amdgcn_wmma_f32_16x16x4_f32(
                false, a, false, b, (short)0, acc[t], false, false);
        }
    }
    const float omb = 1.0f - beta;
    #pragma unroll
    for (int t = 0; t < 6; ++t) {
        const int n = t * 16 + m;
        #pragma unroll
        for (int r = 0; r < 8; ++r) {
            const int row = m0 + r + 8 * g;
            if (row < N) {
                size_t o = (size_t)row * HD + n;
                Y[o] = omb * H[o] + beta * acc[t][r];
            }
        }
    }
}

// ---------------- batchnorm stats: one block per feature ----------------
__global__ __launch_bounds__(256)
void k_bn_stats(const float* __restrict__ H, float* __restrict__ stats, int N) {
    __shared__ float ssum[256];
    __shared__ float ssq[256];
    const int f = blockIdx.x;
    float s = 0.f, s2 = 0.f;
    for (int n = threadIdx.x; n < N; n += 256) {
        float v = H[(size_t)n * HD + f];
        s += v; s2 += v * v;
    }
    ssum[threadIdx.x] = s; ssq[threadIdx.x] = s2;
    __syncthreads();
    for (int off = 128; off > 0; off >>= 1) {
        if (threadIdx.x < off) {
            ssum[threadIdx.x] += ssum[threadIdx.x + off];
            ssq[threadIdx.x]  += ssq[threadIdx.x + off];
        }
        __syncthreads();
    }
    if (threadIdx.x == 0) {
        float mean = ssum[0] / (float)N;
        float var  = ssq[0] / (float)N - mean * mean;
        stats[f]      = mean;
        stats[HD + f] = rsqrtf(var + BN_EPS);
    }
}
__global__ void k_bn_apply_relu(const float* __restrict__ H, const float* __restrict__ stats,
                                const float* __restrict__ gamma, const float* __restrict__ beta,
                                float* __restrict__ Y, int N) {
    long long idx = (long long)blockIdx.x * blockDim.x + threadIdx.x;
    if (idx < (long long)N * HD) {
        int f = (int)(idx % HD);
        float v = gamma[f] * (H[idx] - stats[f]) * stats[HD + f] + beta[f];
        Y[idx] = v > 0.f ? v : 0.f;
    }
}

// ---------------- fc2: OUT = X[N,96] @ W[96,40] + b ----------------
__global__ __launch_bounds__(256)
void k_fc2(const float* __restrict__ X, const float* __restrict__ W,
           const float* __restrict__ bias, float* __restrict__ OUT, int N) {
    __shared__ v2f sW2[48 * 48];                    // 18 KB, padded 40 -> 48 cols
    const int tid  = threadIdx.x;
    const int lane = tid & 31;
    const int m    = lane & 15;
    const int g    = lane >> 4;
    for (int i = tid; i < 48 * 48; i += 256) {
        int p = i / 48, n = i % 48;
        v2f w;
        w.x = (n < OUTD) ? W[(2 * p)     * OUTD + n] : 0.f;
        w.y = (n < OUTD) ? W[(2 * p + 1) * OUTD + n] : 0.f;
        sW2[i] = w;
    }
    __syncthreads();

    int mtiles = (N + 15) >> 4;
    int mt = blockIdx.x * 8 + (tid >> 5);
    if (mt >= mtiles) mt = mtiles - 1;
    const int m0 = mt << 4;
    const int ar = min(m0 + m, N - 1);
    const float* Arow = X + (size_t)ar * HD;

    v8f acc[3];
    #pragma unroll
    for (int t = 0; t < 3; ++t) acc[t] = (v8f){0,0,0,0,0,0,0,0};

    for (int k = 0; k < HD; k += 4) {
        const int kp = (k >> 1) + g;
        v2f a;
        a.x = Arow[k + 2 * g];
        a.y = Arow[k + 2 * g + 1];
        #pragma unroll
        for (int t = 0; t < 3; ++t) {
            v2f b = sW2[kp * 48 + t * 16 + m];
            acc[t] = __builtin_amdgcn_wmma_f32_16x16x4_f32(
                false, a, false, b, (short)0, acc[t], false, false);
        }
    }
    #pragma unroll
    for (int t = 0; t < 3; ++t) {
        const int n = t * 16 + m;
        if (n < OUTD) {
            const float bv = bias[n];
            #pragma unroll
            for (int r = 0; r < 8; ++r) {
                const int row = m0 + r + 8 * g;
                if (row < N) OUT[(size_t)row * OUTD + n] = acc[t][r] + bv;
            }
        }
    }
}

// ---------------- launcher ----------------
extern "C" void kernel_launch(void* const* d_in, const int* in_sizes, int n_in,
                              void* d_out, int out_size, void* d_ws, size_t ws_size,
                              hipStream_t stream) {
    const float* x      = (const float*)d_in[0];
    const int*   ei     = (const int*)  d_in[1];
    const float* fc1_w  = (const float*)d_in[2];
    const float* fc1_b  = (const float*)d_in[3];
    const float* conv_w = (const float*)d_in[4];
    const float* bn_g   = (const float*)d_in[5];
    const float* bn_b   = (const float*)d_in[6];
    const float* fc2_w  = (const float*)d_in[7];
    const float* fc2_b  = (const float*)d_in[8];

    const int N = in_sizes[0] / IND;
    const int E = in_sizes[1] / 2;

    float* ws    = (float*)d_ws;
    float* dinv  = ws;                       // N
    float* x0    = dinv + N;                 // N*96
    float* xa    = x0   + (size_t)N * HD;    // N*96
    float* hb    = xa   + (size_t)N * HD;    // N*96
    float* hc    = hb   + (size_t)N * HD;    // N*96
    float* stats = hc   + (size_t)N * HD;    // 192

    const int mtiles   = (N + 15) / 16;
    const int gemm_blk = (mtiles + 7) / 8;

    k_deg_init <<<(N + 255) / 256, 256, 0, stream>>>(dinv, N);
    k_deg_count<<<(E + 255) / 256, 256, 0, stream>>>(ei + E, dinv, E);
    k_deg_rsqrt<<<(N + 255) / 256, 256, 0, stream>>>(dinv, N);

    k_fc1<<<gemm_blk, 256, 0, stream>>>(x, fc1_w, fc1_b, x0, N);

    const long long NH = (long long)N * HD;
    const int ew_blk = (E + 7) / 8;          // one wave (32 lanes) per edge
    const float* xin = x0;
    for (int i = 0; i < 3; ++i) {
        const float beta = logf(0.5f / (float)(i + 1) + 1.0f);
        k_agg_init <<<(int)((NH + 255) / 256), 256, 0, stream>>>(xin, x0, dinv, hb, N);
        k_agg_edges<<<ew_blk, 256, 0, stream>>>(ei, dinv, xin, hb, E);
        k_conv     <<<gemm_blk, 256, 0, stream>>>(hb, conv_w + (size_t)i * HD * HD, hc, N, beta);
        k_bn_stats <<<HD, 256, 0, stream>>>(hc, stats, N);
        k_bn_apply_relu<<<(int)((NH + 255) / 256), 256, 0, stream>>>(
            hc, stats, bn_g + i * HD, bn_b + i * HD, xa, N);
        xin = xa;
    }

    k_fc2<<<gemm_blk, 256, 0, stream>>>(xin, fc2_w, fc2_b, (float*)d_out, N);
}